// TrustGCN_18330920419684
// MI455X (gfx1250) — compile-verified
//
#include <hip/hip_runtime.h>
#include <hip/hip_bf16.h>
#include <math.h>

typedef __attribute__((ext_vector_type(2))) float v2f;
typedef __attribute__((ext_vector_type(8))) float v8f;

#define BN_EPS 1e-5f
#define TPB 256

__device__ __forceinline__ void atomic_add_f32(float* p, float v) {
  __hip_atomic_fetch_add(p, v, __ATOMIC_RELAXED, __HIP_MEMORY_SCOPE_AGENT);
}

__device__ __forceinline__ float elu1(float x) {
  return x > 0.0f ? x : (__expf(x) - 1.0f);
}

// ---------------- degree / norm ----------------
__global__ void k_deg_init(float* deg, int n) {
  int i = blockIdx.x * blockDim.x + threadIdx.x;
  if (i < n) deg[i] = 1.0f;  // self loop
}
__global__ void k_deg_count(const int* __restrict__ dst, float* deg, int e) {
  int i = blockIdx.x * blockDim.x + threadIdx.x;
  if (i < e) atomic_add_f32(&deg[dst[i]], 1.0f);
}
__global__ void k_deg_rsqrt(float* deg, int n) {
  int i = blockIdx.x * blockDim.x + threadIdx.x;
  if (i < n) deg[i] = rsqrtf(deg[i]);   // deg >= 1 always (self loops)
}

// ---------------- WMMA f32 GEMM: H[n,COUT] = X[n,CIN] @ W[CIN,COUT] ----------------
// One wave computes a 16-node row tile across all COUT columns with
// V_WMMA_F32_16X16X4_F32, K accumulated 4 at a time. All (CIN/4)*(COUT/16)
// B-fragments are preloaded into registers before the tile loop (W is tiny and
// tile-invariant). Only full 16-row tiles are processed here (no predication:
// EXEC stays all-ones, stores are straight-line); tail rows go to k_gemm_tail.
// A frag (16x4 f32): lanes 0-15 rows M=0..15 K={0,1}; lanes 16-31 K={2,3}.
// B frag (4x16 f32): lane%16 = column; lane/16 selects K pair, 2 K-rows/VGPR.
// C/D (16x16 f32): vgpr r, lane l -> row = r + 8*(l/16), col = l%16.
template <int CIN, int COUT, bool FUSE_BIAS_ELU>
__global__ void k_gemm_wmma(const float* __restrict__ X, const float* __restrict__ W,
                            const float* __restrict__ bias, float* __restrict__ H, int n) {
  constexpr int KS = CIN / 4;
  constexpr int NT = COUT / 16;
  const int lane = threadIdx.x & 31;
  const int half = lane >> 4;   // 0 or 1
  const int l16  = lane & 15;
  const int wave   = (blockIdx.x * blockDim.x + threadIdx.x) >> 5;
  const int nwaves = (gridDim.x * blockDim.x) >> 5;
  const int ntiles = n >> 4;    // full tiles only

  // Preload all B fragments (tile-invariant).
  v2f bf[KS][NT];
#pragma unroll
  for (int k = 0; k < KS; ++k) {
#pragma unroll
    for (int j = 0; j < NT; ++j) {
      const int col = j * 16 + l16;
      bf[k][j].x = W[(k * 4 + half * 2 + 0) * COUT + col];
      bf[k][j].y = W[(k * 4 + half * 2 + 1) * COUT + col];
    }
  }

  float bcol[FUSE_BIAS_ELU ? NT : 1];
  if constexpr (FUSE_BIAS_ELU) {
#pragma unroll
    for (int j = 0; j < NT; ++j) bcol[j] = bias[j * 16 + l16];
  }

  for (int t = wave; t < ntiles; t += nwaves) {      // wave-uniform control flow
    const int row0 = t << 4;
    const int ra = row0 + l16;
    v8f acc[NT] = {};
#pragma unroll
    for (int k = 0; k < KS; ++k) {
      v2f a;
      a.x = X[(size_t)ra * CIN + k * 4 + half * 2 + 0];
      a.y = X[(size_t)ra * CIN + k * 4 + half * 2 + 1];
#pragma unroll
      for (int j = 0; j < NT; ++j) {
        acc[j] = __builtin_amdgcn_wmma_f32_16x16x4_f32(
            false, a, false, bf[k][j], (short)0, acc[j], false, false);
      }
    }
#pragma unroll
    for (int j = 0; j < NT; ++j) {
      const int col = j * 16 + l16;
#pragma unroll
      for (int r = 0; r < 8; ++r) {
        float v = acc[j][r];
        if constexpr (FUSE_BIAS_ELU) v = elu1(v + bcol[j]);
        H[(size_t)(row0 + half * 8 + r) * COUT + col] = v;
      }
    }
  }
}

// Scalar fallback for tail rows [n & ~15, n); launched only when n % 16 != 0.
template <int CIN, int COUT, bool FUSE_BIAS_ELU>
__global__ void k_gemm_tail(const float* __restrict__ X, const float* __restrict__ W,
                            const float* __restrict__ bias, float* __restrict__ H,
                            int row_lo, int n) {
  const int i = blockIdx.x * blockDim.x + threadIdx.x;
  const int row = row_lo + i / COUT;
  const int col = i % COUT;
  if (row >= n) return;
  float acc = 0.f;
#pragma unroll
  for (int k = 0; k < CIN; ++k) acc = fmaf(X[(size_t)row * CIN + k], W[k * COUT + col], acc);
  if constexpr (FUSE_BIAS_ELU) acc = elu1(acc + bias[col]);
  H[(size_t)row * COUT + col] = acc;
}

// ---------------- self-loop seed: AGG = H * dinv^2 ----------------
template <int C>
__global__ void k_selfloop(const float* __restrict__ H, const float* __restrict__ dinv,
                           float* __restrict__ AGG, int n) {
  const int total = n * C;
  for (int i = blockIdx.x * blockDim.x + threadIdx.x; i < total;
       i += gridDim.x * blockDim.x) {
    const float di = dinv[i / C];
    AGG[i] = H[i] * di * di;
  }
}

// ---------------- edge scatter: AGG[d] += H[s] * dinv[s]*dinv[d] ----------------
template <int C>
__global__ void k_scatter(const int* __restrict__ src, const int* __restrict__ dst,
                          const float* __restrict__ dinv, const float* __restrict__ H,
                          float* __restrict__ AGG, int e) {
  const int total = e * C;
  for (int i = blockIdx.x * blockDim.x + threadIdx.x; i < total;
       i += gridDim.x * blockDim.x) {
    const int ed = i / C;           // C is a power of two -> shift
    const int c  = i & (C - 1);
    const int s = src[ed], d = dst[ed];
    const float nrm = dinv[s] * dinv[d];
    atomic_add_f32(&AGG[(size_t)d * C + c], H[(size_t)s * C + c] * nrm);
  }
}

__global__ void k_zero_stats(float* stats) {
  if (threadIdx.x < 128) stats[threadIdx.x] = 0.0f;  // gsum[64] ++ gsq[64]
}

// ---------------- VAL = relu(AGG + bias); accumulate per-channel sum / sumsq ----------------
template <int C>
__global__ void k_relu_stats(const float* __restrict__ AGG, const float* __restrict__ bias,
                             float* __restrict__ VAL, float* __restrict__ gsum,
                             float* __restrict__ gsq, int n) {
  __shared__ float s_sum[TPB];
  __shared__ float s_sq[TPB];
  const int tid = threadIdx.x;
  const int c = tid & (C - 1);
  const int rpb = TPB / C;
  const int rloc = tid / C;
  float sum = 0.f, sq = 0.f;
  for (int row = blockIdx.x * rpb + rloc; row < n; row += gridDim.x * rpb) {
    float v = AGG[(size_t)row * C + c] + bias[c];
    v = v > 0.f ? v : 0.f;
    VAL[(size_t)row * C + c] = v;
    sum += v;
    sq  += v * v;
  }
  s_sum[tid] = sum;
  s_sq[tid]  = sq;
  __syncthreads();
  for (int str = TPB / 2; str >= C; str >>= 1) {
    if (tid < str) {
      s_sum[tid] += s_sum[tid + str];
      s_sq[tid]  += s_sq[tid + str];
    }
    __syncthreads();
  }
  if (tid < C) {
    atomic_add_f32(&gsum[tid], s_sum[tid]);
    atomic_add_f32(&gsq[tid],  s_sq[tid]);
  }
}

// ---------------- OUT = elu((VAL - m) * rsqrt(var+eps) * g + bt) ----------------
template <int C>
__global__ void k_bn_elu(const float* __restrict__ VAL, const float* __restrict__ gsum,
                         const float* __restrict__ gsq, const float* __restrict__ gamma,
                         const float* __restrict__ beta, float* __restrict__ OUT, int n) {
  const int total = n * C;
  const float invn = 1.0f / (float)n;
  for (int i = blockIdx.x * blockDim.x + threadIdx.x; i < total;
       i += gridDim.x * blockDim.x) {
    const int c = i & (C - 1);
    const float m = gsum[c] * invn;
    const float var = gsq[c] * invn - m * m;
    const float xh = (VAL[i] - m) * rsqrtf(var + BN_EPS) * gamma[c] + beta[c];
    OUT[i] = elu1(xh);
  }
}

// ---------------- head: z = H[16] @ Wf3[16,2] + b; log_softmax ----------------
__global__ void k_head(const float* __restrict__ H, const float* __restrict__ W,
                       const float* __restrict__ b, float* __restrict__ out, int n) {
  for (int i = blockIdx.x * blockDim.x + threadIdx.x; i < n;
       i += gridDim.x * blockDim.x) {
    float z0 = b[0], z1 = b[1];
#pragma unroll
    for (int k = 0; k < 16; ++k) {
      const float h = H[(size_t)i * 16 + k];
      z0 = fmaf(h, W[k * 2 + 0], z0);
      z1 = fmaf(h, W[k * 2 + 1], z1);
    }
    const float mx = fmaxf(z0, z1);
    const float l = __logf(__expf(z0 - mx) + __expf(z1 - mx));
    out[(size_t)i * 2 + 0] = z0 - mx - l;
    out[(size_t)i * 2 + 1] = z1 - mx - l;
  }
}

static inline int grid_for(long long total) {
  long long b = (total + TPB - 1) / TPB;
  if (b > 262144) b = 262144;
  if (b < 1) b = 1;
  return (int)b;
}

template <int CIN, int COUT, bool FUSE>
static inline void launch_gemm(const float* X, const float* W, const float* bias,
                               float* H, int n, hipStream_t stream) {
  const int ntiles = n >> 4;
  const int g = grid_for((long long)ntiles * 32);
  k_gemm_wmma<CIN, COUT, FUSE><<<g, TPB, 0, stream>>>(X, W, bias, H, n);
  const int tail = n & 15;
  if (tail) {  // deterministic: depends only on input sizes (tail==0 for N=100000)
    const int tg = (tail * COUT + TPB - 1) / TPB;
    k_gemm_tail<CIN, COUT, FUSE><<<tg, TPB, 0, stream>>>(X, W, bias, H, n & ~15, n);
  }
}

extern "C" void kernel_launch(void* const* d_in, const int* in_sizes, int n_in,
                              void* d_out, int out_size, void* d_ws, size_t ws_size,
                              hipStream_t stream) {
  const int n = in_sizes[0] / 4;   // N nodes (x is [N,4])
  const int e = in_sizes[1] / 2;   // E edges (edge_index is [2,E])

  const float* x  = (const float*)d_in[0];
  const int* ei   = (const int*)d_in[1];
  const int* src  = ei;
  const int* dst  = ei + e;
  const float* W1 = (const float*)d_in[2];  const float* b1 = (const float*)d_in[3];
  const float* g1 = (const float*)d_in[4];  const float* t1 = (const float*)d_in[5];
  const float* W2 = (const float*)d_in[6];  const float* b2 = (const float*)d_in[7];
  const float* g2 = (const float*)d_in[8];  const float* t2 = (const float*)d_in[9];
  const float* W3 = (const float*)d_in[10]; const float* b3 = (const float*)d_in[11];
  const float* g3 = (const float*)d_in[12]; const float* t3 = (const float*)d_in[13];
  const float* Wf1 = (const float*)d_in[14]; const float* bf1 = (const float*)d_in[15];
  const float* Wf2 = (const float*)d_in[16]; const float* bf2 = (const float*)d_in[17];
  const float* Wf3 = (const float*)d_in[18]; const float* bf3 = (const float*)d_in[19];

  // workspace: 3 rotating N*64 buffers + dinv[N] + stats[128]  (~77.2 MB)
  const size_t n64 = (size_t)n * 64;
  float* B0   = (float*)d_ws;
  float* B1   = B0 + n64;
  float* B2   = B1 + n64;
  float* dinv = B2 + n64;
  float* gsum = dinv + n;
  float* gsq  = gsum + 64;

  const int gN     = grid_for(n);
  const int gE     = grid_for(e);
  const int gStats = 1024;

  // degree -> dinv
  k_deg_init<<<gN, TPB, 0, stream>>>(dinv, n);
  k_deg_count<<<gE, TPB, 0, stream>>>(dst, dinv, e);
  k_deg_rsqrt<<<gN, TPB, 0, stream>>>(dinv, n);

  // ---- GCN layer 1: 4 -> 16 (in: x, H: B0, AGG/out: B1) ----
  launch_gemm<4, 16, false>(x, W1, nullptr, B0, n, stream);
  k_selfloop<16><<<grid_for((long long)n * 16), TPB, 0, stream>>>(B0, dinv, B1, n);
  k_scatter<16><<<grid_for((long long)e * 16), TPB, 0, stream>>>(src, dst, dinv, B0, B1, e);
  k_zero_stats<<<1, 128, 0, stream>>>(gsum);
  k_relu_stats<16><<<gStats, TPB, 0, stream>>>(B1, b1, B0, gsum, gsq, n);
  k_bn_elu<16><<<grid_for((long long)n * 16), TPB, 0, stream>>>(B0, gsum, gsq, g1, t1, B1, n);

  // ---- GCN layer 2: 16 -> 32 (in: B1, H: B0, AGG/out: B2) ----
  launch_gemm<16, 32, false>(B1, W2, nullptr, B0, n, stream);
  k_selfloop<32><<<grid_for((long long)n * 32), TPB, 0, stream>>>(B0, dinv, B2, n);
  k_scatter<32><<<grid_for((long long)e * 32), TPB, 0, stream>>>(src, dst, dinv, B0, B2, e);
  k_zero_stats<<<1, 128, 0, stream>>>(gsum);
  k_relu_stats<32><<<gStats, TPB, 0, stream>>>(B2, b2, B0, gsum, gsq, n);
  k_bn_elu<32><<<grid_for((long long)n * 32), TPB, 0, stream>>>(B0, gsum, gsq, g2, t2, B2, n);

  // ---- GCN layer 3: 32 -> 64 (in: B2, H: B0, AGG/out: B1) ----
  launch_gemm<32, 64, false>(B2, W3, nullptr, B0, n, stream);
  k_selfloop<64><<<grid_for((long long)n * 64), TPB, 0, stream>>>(B0, dinv, B1, n);
  k_scatter<64><<<grid_for((long long)e * 64), TPB, 0, stream>>>(src, dst, dinv, B0, B1, e);
  k_zero_stats<<<1, 128, 0, stream>>>(gsum);
  k_relu_stats<64><<<gStats, TPB, 0, stream>>>(B1, b3, B0, gsum, gsq, n);
  k_bn_elu<64><<<grid_for((long long)n * 64), TPB, 0, stream>>>(B0, gsum, gsq, g3, t3, B1, n);

  // ---- MLP head: 64 -> 32 -> 16 -> 2 + log_softmax ----
  launch_gemm<64, 32, true>(B1, Wf1, bf1, B2, n, stream);
  launch_gemm<32, 16, true>(B2, Wf2, bf2, B0, n, stream);
  k_head<<<gN, TPB, 0, stream>>>(B0, Wf3, bf3, (float*)d_out, n);
}